// BilinearInteractionLayer_59485297050179
// MI455X (gfx1250) — compile-verified
//
#include <hip/hip_runtime.h>
#include <stdint.h>

typedef __attribute__((ext_vector_type(2))) float v2f;
typedef __attribute__((ext_vector_type(8))) float v8f;

constexpr int BB       = 2048;            // batch
constexpr int F        = 32;              // features
constexpr int D        = 64;              // embedding dim
constexpr int P        = F * (F - 1) / 2; // 496 pairs
constexpr int OUTW     = P * D;           // 31744 output cols
constexpr int ROWS     = 128;             // batch rows per block
constexpr int V_STRIDE = 68;              // padded LDS row stride for Vi (floats)

// Swizzled W layout in LDS: float offset = kc*320 + h*160 + e*2 + q
// holds W[4*kc + 2*h + q][e]   (kc = K-chunk 0..15, h = lane-half, q = pair pos)
// -> each lane's B fragment {W[k][c], W[k+1][c]} is one aligned float2,
//    and the two lane halves read complementary 32-bank sets (conflict-free).

__global__ __launch_bounds__(256)
void bilinear_pair_wmma_f32(const float* __restrict__ inp,   // (B, F, D)
                            const float* __restrict__ Wt,    // (P, D, D)
                            float* __restrict__ out)         // (B, P*D)
{
    __shared__ float sW[16 * 320];          // 20 KB, swizzled
    __shared__ float sVi[ROWS * V_STRIDE];  // 34 KB, padded rows

    const int p  = blockIdx.y;
    const int b0 = blockIdx.x * ROWS;

    // decode upper-triangular pair index p -> (i, j)
    int i = 0, rem = p;
    while (rem >= F - 1 - i) { rem -= F - 1 - i; ++i; }
    const int j = i + 1 + rem;

    const int tid = threadIdx.x;

    // ---- stage Vi tile (128 rows x 64 f32) via async global->LDS DMA ----
    // Straight copy with per-lane padded LDS addresses; 16B-aligned both sides.
    {
        #pragma unroll
        for (int t = 0; t < 8; ++t) {
            int idx = tid + t * 256;          // 0..2047 float4s
            int row = idx >> 4, c4 = idx & 15;
            uint32_t lds = (uint32_t)(uintptr_t)&sVi[row * V_STRIDE + c4 * 4];
            const float* g = inp + ((size_t)(b0 + row) * F + i) * D + c4 * 4;
            asm volatile("global_load_async_to_lds_b128 %0, %1, off"
                         :: "v"(lds), "v"(g) : "memory");
        }
    }

    // ---- stage W[p] into swizzled LDS layout (register pass-through) ----
    {
        const float4* src = (const float4*)(Wt + (size_t)p * D * D);
        #pragma unroll
        for (int t = 0; t < 2; ++t) {
            int idx = tid + t * 256;          // 0..511 : (row-pair, c4)
            int pr = idx >> 4, c4 = idx & 15; // pr = 0..31 pairs of rows
            float4 ra = src[(2 * pr + 0) * 16 + c4];   // row k   = 2*pr
            float4 rb = src[(2 * pr + 1) * 16 + c4];   // row k+1
            const int kc = pr >> 1, h = pr & 1;
            float* dst = &sW[kc * 320 + h * 160 + (c4 * 4) * 2];
            v2f p0 = {ra.x, rb.x}; ((v2f*)dst)[0] = p0;
            v2f p1 = {ra.y, rb.y}; ((v2f*)dst)[1] = p1;
            v2f p2 = {ra.z, rb.z}; ((v2f*)dst)[2] = p2;
            v2f p3 = {ra.w, rb.w}; ((v2f*)dst)[3] = p3;
        }
    }

    // all async copies from this wave landed in LDS, then sync the block
    asm volatile("s_wait_asynccnt 0x0" ::: "memory");
    __syncthreads();

    const int wave = tid >> 5;   // 0..7 -> 16-row batch slice
    const int lane = tid & 31;
    const int half = lane >> 4;  // 0: K 0/1, 1: K 2/3 (ISA A/B half-split-K layout)
    const int lrow = lane & 15;  // M for A, N for B/C/D

    const float* aBase = &sVi[(wave * 16 + lrow) * V_STRIDE + 2 * half];
    const float* bBase = &sW[half * 160 + lrow * 2];

    v8f acc0 = {}, acc1 = {}, acc2 = {}, acc3 = {};

    #pragma unroll
    for (int kc = 0; kc < 16; ++kc) {
        v2f a;
        a.x = aBase[kc * 4];
        a.y = aBase[kc * 4 + 1];
        const v2f* bp = (const v2f*)(bBase + kc * 320);
        v2f b0 = bp[0];    // N-tile 0 (n stride = 32 floats = 16 v2f)
        v2f b1 = bp[16];   // N-tile 1
        v2f b2 = bp[32];   // N-tile 2
        v2f b3 = bp[48];   // N-tile 3
        acc0 = __builtin_amdgcn_wmma_f32_16x16x4_f32(false, a, false, b0, (short)0, acc0, false, false);
        acc1 = __builtin_amdgcn_wmma_f32_16x16x4_f32(false, a, false, b1, (short)0, acc1, false, false);
        acc2 = __builtin_amdgcn_wmma_f32_16x16x4_f32(false, a, false, b2, (short)0, acc2, false, false);
        acc3 = __builtin_amdgcn_wmma_f32_16x16x4_f32(false, a, false, b3, (short)0, acc3, false, false);
    }

    // ---- epilogue: multiply by v_j (read-once from global) and store ----
    #pragma unroll
    for (int r = 0; r < 8; ++r) {
        const int m   = r + 8 * half;    // C/D layout: VGPR r holds M = r (+8 upper half)
        const int row = b0 + wave * 16 + m;
        const float* vjRow = inp + ((size_t)row * F + j) * D;
        float* oRow = out + (size_t)row * OUTW + (size_t)p * D;
        oRow[ 0 + lrow] = acc0[r] * vjRow[ 0 + lrow];
        oRow[16 + lrow] = acc1[r] * vjRow[16 + lrow];
        oRow[32 + lrow] = acc2[r] * vjRow[32 + lrow];
        oRow[48 + lrow] = acc3[r] * vjRow[48 + lrow];
    }
}

extern "C" void kernel_launch(void* const* d_in, const int* in_sizes, int n_in,
                              void* d_out, int out_size, void* d_ws, size_t ws_size,
                              hipStream_t stream) {
    (void)in_sizes; (void)n_in; (void)out_size; (void)d_ws; (void)ws_size;
    const float* inp = (const float*)d_in[0];   // (2048, 32, 64) f32
    const float* Wt  = (const float*)d_in[1];   // (496, 64, 64) f32
    float* out = (float*)d_out;                 // (2048, 31744) f32

    dim3 grid(BB / ROWS, P);                    // (16, 496)
    bilinear_pair_wmma_f32<<<grid, 256, 0, stream>>>(inp, Wt, out);
}